// OptNetDenoiser_15272903704595
// MI455X (gfx1250) — compile-verified
//
#include <hip/hip_runtime.h>

typedef __attribute__((ext_vector_type(2))) float v2f;
typedef __attribute__((ext_vector_type(8))) float v8f;

#define T_DIM   128
#define M_CON   127
#define NITER   30
#define SIGMA   0.1f
#define LDA     129          // padded row stride (odd -> conflict-free column access)
#define THREADS 256
#define NB      16           // Cholesky block size
#define NBLK    8            // 128 / 16

// LDS float offsets
#define OFF_AS  0
#define OFF_K   (128 * LDA)
#define OFF_VEC (2 * 128 * LDA)
#define SMEM_FLOATS (2 * 128 * LDA + 1544)

// Wave-local LDS sync: DS ops of one wave complete in order; waiting for
// DScnt==0 makes prior LDS stores visible to subsequent loads from any lane
// of the same (lockstep) wave32 -- no workgroup barrier needed.
static __device__ __forceinline__ void wave_lds_sync() {
    asm volatile("s_wait_dscnt 0x0" ::: "memory");
}

__launch_bounds__(THREADS)
__global__ void optnet_ipm_kernel(const float* __restrict__ y,
                                  const float* __restrict__ G,
                                  const float* __restrict__ theta,
                                  float* __restrict__ out)
{
    extern __shared__ float smem[];
    float* As  = smem + OFF_AS;       // 128 x LDA : sqrt(d) * G (row 127 zero)
    float* K   = smem + OFF_K;        // 128 x LDA : K -> Cholesky factor L (lower)
    float* vec = smem + OFF_VEC;
    float* yv    = vec;               // 128
    float* zv    = vec + 128;
    float* rhsv  = vec + 256;         // rhs -> u -> dz in place
    float* lamv  = vec + 384;
    float* sv    = vec + 512;
    float* sdv   = vec + 640;         // sqrt(lam/s)
    float* rpv   = vec + 768;
    float* rcv   = vec + 896;
    float* dsv   = vec + 1024;
    float* dlamv = vec + 1152;
    float* wv    = vec + 1280;
    float* redv  = vec + 1408;
    float* scal  = vec + 1536;        // [0]=mu [1]=alpha [2]=h

    const int tid = threadIdx.x;
    const int b   = blockIdx.x;

    if (tid < T_DIM) {
        zv[tid]   = 0.0f;
        yv[tid]   = y[b * T_DIM + tid];
        lamv[tid] = 1.0f;
        sv[tid]   = 1.0f;
    }
    if (tid == 0) scal[2] = theta[0];
    __syncthreads();
    const float hval = scal[2];

    const int wid  = tid >> 5;
    const int lane = tid & 31;
    const int lrow = lane & 15;
    const int lhi  = lane >> 4;      // A/B frag: 0 -> K=0,1 ; 1 -> K=2,3

    for (int it = 0; it < NITER; ++it) {
        // ================= scaling, complementarity =================
        if (tid < M_CON) {
            float l = lamv[tid], ss = sv[tid];
            redv[tid] = l * ss;
            sdv[tid]  = sqrtf(l / ss);
        } else if (tid < 128) {
            sdv[tid]  = 1.0f;
            redv[tid] = 0.0f;
        }
        __syncthreads();
        if (wid == 0) {                       // wave32 shuffle reduction for mu
            float acc = redv[lane] + redv[lane + 32] + redv[lane + 64] + redv[lane + 96];
            acc += __shfl_xor(acc, 16, 32);
            acc += __shfl_xor(acc,  8, 32);
            acc += __shfl_xor(acc,  4, 32);
            acc += __shfl_xor(acc,  2, 32);
            acc += __shfl_xor(acc,  1, 32);
            if (lane == 0) scal[0] = acc * (1.0f / (float)M_CON);
        }
        __syncthreads();
        const float mu = scal[0];
        if (tid < M_CON) rcv[tid] = sv[tid] * lamv[tid] - SIGMA * mu;

        // ================= As = sqrt(d) .* G =================
        for (int idx = tid; idx < M_CON * T_DIM; idx += THREADS) {
            int m = idx >> 7;
            int t = idx & 127;
            As[m * LDA + t] = sdv[m] * G[idx];
        }
        if (tid < T_DIM) As[M_CON * LDA + tid] = 0.0f;
        __syncthreads();

        // ======== rp[m] = z.G_m + s - h ; w[m] = (lam + (lam*rp-rc)/s)/sd ========
        if (tid < M_CON) {
            float dot = 0.0f;
            #pragma unroll 8
            for (int t = 0; t < T_DIM; ++t) dot += zv[t] * As[tid * LDA + t];
            float rp = dot / sdv[tid] + sv[tid] - hval;
            rpv[tid] = rp;
            float cf = (lamv[tid] * rp - rcv[tid]) / sv[tid];
            wv[tid]  = (lamv[tid] + cf) / sdv[tid];
        }
        __syncthreads();

        // ======== rhs[t] = -(z - y + sum_m w[m] * As[m,t]) ========
        if (tid < T_DIM) {
            float acc = 0.0f;
            #pragma unroll 8
            for (int m = 0; m < M_CON; ++m) acc += wv[m] * As[m * LDA + tid];
            rhsv[tid] = -(zv[tid] - yv[tid] + acc);
        }
        __syncthreads();

        // ======== K = I + As^T As : lower-triangle 16x16 tiles via WMMA ========
        for (int q = wid; q < 36; q += 8) {
            int ti = 0;
            while ((ti + 1) * (ti + 2) / 2 <= q) ++ti;
            int tj = q - ti * (ti + 1) / 2;       // tj <= ti (lower triangle)
            const int t0 = ti * 16, s0 = tj * 16;
            v8f acc = {};
            #pragma unroll 4
            for (int m0 = 0; m0 < 128; m0 += 4) {
                const int mA = m0 + lhi * 2;
                v2f a, bf;
                a.x  = As[mA * LDA + t0 + lrow];
                a.y  = As[(mA + 1) * LDA + t0 + lrow];
                bf.x = As[mA * LDA + s0 + lrow];
                bf.y = As[(mA + 1) * LDA + s0 + lrow];
                acc = __builtin_amdgcn_wmma_f32_16x16x4_f32(
                        false, a, false, bf, (short)0, acc, false, false);
            }
            if (ti == tj) {                       // identity only on diagonal tiles
                #pragma unroll
                for (int v = 0; v < 8; ++v)
                    acc[v] += ((v + lhi * 8) == lrow) ? 1.0f : 0.0f;
            }
            #pragma unroll
            for (int v = 0; v < 8; ++v)
                K[(t0 + v + lhi * 8) * LDA + s0 + lrow] = acc[v];
        }
        __syncthreads();

        // ======== blocked Cholesky: single-wave panel + WMMA trailing update ========
        for (int jb = 0; jb < NBLK; ++jb) {
            // ---- panel factorization (columns jb*16 .. jb*16+15, all rows below)
            // done by wave 0 alone: lockstep wave32, wave-local LDS sync only.
            if (wid == 0) {
                const int kend = jb * NB + NB;
                for (int c = 0; c < NB; ++c) {
                    const int j = jb * NB + c;
                    float dval = K[j * LDA + j];       // broadcast read
                    float ljj  = sqrtf(dval);
                    float inv  = 1.0f / ljj;
                    if (lane == 0) K[j * LDA + j] = ljj;
                    for (int i = j + 1 + lane; i < T_DIM; i += 32)
                        K[i * LDA + j] *= inv;
                    wave_lds_sync();
                    for (int i = j + 1 + lane; i < T_DIM; i += 32) {
                        const float lij = K[i * LDA + j];
                        for (int k = j + 1; k < kend; ++k)
                            K[i * LDA + k] -= lij * K[k * LDA + j];
                    }
                    wave_lds_sync();
                }
            }
            __syncthreads();

            // ---- trailing update K[ib][kb] -= L[ib]*L[kb]^T via WMMA (8 waves)
            const int nb = NBLK - 1 - jb;
            const int ntiles = nb * (nb + 1) / 2;
            const int p0 = jb * NB;
            for (int q = wid; q < ntiles; q += 8) {
                int a = 0;
                while ((a + 1) * (a + 2) / 2 <= q) ++a;
                int bc = q - a * (a + 1) / 2;
                const int r0 = (jb + 1 + a) * NB;
                const int c0 = (jb + 1 + bc) * NB;
                v8f acc;
                #pragma unroll
                for (int v = 0; v < 8; ++v)
                    acc[v] = K[(r0 + v + lhi * 8) * LDA + c0 + lrow];
                #pragma unroll
                for (int s = 0; s < NB; s += 4) {
                    const int kf = p0 + s + lhi * 2;
                    v2f a2, b2;
                    a2.x = -K[(r0 + lrow) * LDA + kf];      // negate A -> C - A*B^T
                    a2.y = -K[(r0 + lrow) * LDA + kf + 1];
                    b2.x =  K[(c0 + lrow) * LDA + kf];      // B = L[kb] read transposed
                    b2.y =  K[(c0 + lrow) * LDA + kf + 1];
                    acc = __builtin_amdgcn_wmma_f32_16x16x4_f32(
                            false, a2, false, b2, (short)0, acc, false, false);
                }
                #pragma unroll
                for (int v = 0; v < 8; ++v)
                    K[(r0 + v + lhi * 8) * LDA + c0 + lrow] = acc[v];
            }
            __syncthreads();
        }

        // ======== blocked forward solve L u = rhs ========
        for (int jb = 0; jb < NBLK; ++jb) {
            const int j0 = jb * NB;
            if (wid == 0) {
                for (int c = 0; c < NB; ++c) {
                    const int j = j0 + c;
                    if (lane == c) rhsv[j] /= K[j * LDA + j];
                    wave_lds_sync();
                    const float xj = rhsv[j];
                    if (lane > c && lane < NB)
                        rhsv[j0 + lane] -= K[(j0 + lane) * LDA + j] * xj;
                    wave_lds_sync();
                }
            }
            __syncthreads();
            for (int i = j0 + NB + tid; i < T_DIM; i += THREADS) {
                float acc = 0.0f;
                #pragma unroll
                for (int c = 0; c < NB; ++c)
                    acc += K[i * LDA + j0 + c] * rhsv[j0 + c];
                rhsv[i] -= acc;
            }
            __syncthreads();
        }

        // ======== blocked back solve L^T dz = u (rhsv becomes dz) ========
        for (int jb = NBLK - 1; jb >= 0; --jb) {
            const int j0 = jb * NB;
            if (wid == 0) {
                for (int c = NB - 1; c >= 0; --c) {
                    const int j = j0 + c;
                    if (lane == c) rhsv[j] /= K[j * LDA + j];
                    wave_lds_sync();
                    const float xj = rhsv[j];
                    if (lane < c)
                        rhsv[j0 + lane] -= K[j * LDA + j0 + lane] * xj;
                    wave_lds_sync();
                }
            }
            __syncthreads();
            for (int i = tid; i < j0; i += THREADS) {
                float acc = 0.0f;
                #pragma unroll
                for (int c = 0; c < NB; ++c)
                    acc += K[(j0 + c) * LDA + i] * rhsv[j0 + c];
                rhsv[i] -= acc;
            }
            __syncthreads();
        }

        // ======== ds, dlam, fraction-to-boundary, update ========
        if (tid < M_CON) {
            float dot = 0.0f;
            #pragma unroll 8
            for (int t = 0; t < T_DIM; ++t) dot += rhsv[t] * As[tid * LDA + t];
            float ds = -rpv[tid] - dot / sdv[tid];
            dsv[tid] = ds;
            float dl = (-rcv[tid] - lamv[tid] * ds) / sv[tid];
            dlamv[tid] = dl;
            float r1 = (ds < 0.0f) ? (-sv[tid] / ds) : 1e9f;
            float r2 = (dl < 0.0f) ? (-lamv[tid] / dl) : 1e9f;
            redv[tid] = fminf(r1, r2);
        } else if (tid < 128) {
            redv[tid] = 1e9f;
        }
        __syncthreads();
        if (wid == 0) {                        // wave32 shuffle min-reduction for alpha
            float a = fminf(fminf(redv[lane], redv[lane + 32]),
                            fminf(redv[lane + 64], redv[lane + 96]));
            a = fminf(a, __shfl_xor(a, 16, 32));
            a = fminf(a, __shfl_xor(a,  8, 32));
            a = fminf(a, __shfl_xor(a,  4, 32));
            a = fminf(a, __shfl_xor(a,  2, 32));
            a = fminf(a, __shfl_xor(a,  1, 32));
            if (lane == 0) scal[1] = fminf(1.0f, 0.99f * a);
        }
        __syncthreads();
        const float alpha = scal[1];
        if (tid < T_DIM) zv[tid] += alpha * rhsv[tid];
        if (tid < M_CON) {
            lamv[tid] += alpha * dlamv[tid];
            sv[tid]   += alpha * dsv[tid];
        }
        __syncthreads();
    }

    if (tid < T_DIM) out[b * T_DIM + tid] = zv[tid];
}

extern "C" void kernel_launch(void* const* d_in, const int* in_sizes, int n_in,
                              void* d_out, int out_size, void* d_ws, size_t ws_size,
                              hipStream_t stream) {
    const float* y     = (const float*)d_in[0];
    const float* G     = (const float*)d_in[1];
    const float* theta = (const float*)d_in[2];
    float* out = (float*)d_out;

    const int B = in_sizes[0] / T_DIM;           // 2048
    const size_t smem_bytes = (size_t)SMEM_FLOATS * sizeof(float);  // ~138 KB

    optnet_ipm_kernel<<<dim3(B), dim3(THREADS), smem_bytes, stream>>>(y, G, theta, out);
}